// RoIModel_22823456211274
// MI455X (gfx1250) — compile-verified
//
#include <hip/hip_runtime.h>
#include <hip/hip_bf16.h>
#include <stdint.h>

#define B_ 16
#define C_ 80
#define H_ 128
#define W_ 128
#define K_ 100
#define HW_ (H_ * W_)
#define NBINS 4096
#define CANDMAX 2048
#define DOWN_ 4.0f

typedef unsigned int u32x4 __attribute__((ext_vector_type(4)));
typedef int          i32x8 __attribute__((ext_vector_type(8)));
typedef int          i32x4 __attribute__((ext_vector_type(4)));

// ---------- order-preserving float <-> uint key ----------
__device__ __forceinline__ unsigned flip_key(float x) {
  unsigned u = __float_as_uint(x);
  return (u & 0x80000000u) ? ~u : (u | 0x80000000u);
}
__device__ __forceinline__ float unflip_key(unsigned k) {
  unsigned u = (k & 0x80000000u) ? (k & 0x7FFFFFFFu) : ~k;
  return __uint_as_float(u);
}

__device__ __forceinline__ unsigned lds_offset_of(const void* p) {
  return (unsigned)(unsigned long long)(__attribute__((address_space(3))) const char*)p;
}

// ---------- CDNA5 async global -> LDS copy (ASYNCcnt path) ----------
__device__ __forceinline__ void async_load_b128(float* lds_dst, const float* gsrc) {
  unsigned loff = lds_offset_of(lds_dst);
  unsigned long long ga = (unsigned long long)(uintptr_t)gsrc;
  asm volatile("global_load_async_to_lds_b128 %0, %1, off"
               :: "v"(loff), "v"(ga) : "memory");
}
__device__ __forceinline__ void wait_async0() {
  asm volatile("s_wait_asynccnt 0x0" ::: "memory");
}

// Stage rows [r0-1, r0+32] of a 128x128 channel image into a 34x128 LDS tile
// using per-lane async b128 loads. Tile row t <-> global row (r0-1+t).
__device__ __forceinline__ void stage_tile_async(float* tile, const float* img,
                                                 int r0, int tid) {
  for (int it = tid; it < 34 * 32; it += 256) {
    int t = it >> 5;
    int c4 = (it & 31) << 2;          // column in floats, 16B granules
    int gr = r0 - 1 + t;
    if (gr >= 0 && gr < H_)
      async_load_b128(&tile[t * W_ + c4], img + gr * W_ + c4);
  }
  wait_async0();
  __syncthreads();
}

// ---------- CDNA5 Tensor Data Mover (TENSORcnt path) ----------
// 2D tile: nrows x 128 floats, contiguous rows (stride 128), global -> LDS.
// D# layout per CDNA5 ISA.
#if __has_builtin(__builtin_amdgcn_tensor_load_to_lds)
#define HAVE_TDM 1
__device__ __forceinline__ void tdm_load_rows(float* lds_dst, const float* gsrc,
                                              int nrows) {
  unsigned lds_off = lds_offset_of(lds_dst);
  unsigned long long ga = (unsigned long long)(uintptr_t)gsrc;
  unsigned nr = (unsigned)nrows & 0xFFFFu;

  u32x4 g0;
  g0[0] = 1u;                                          // count=1 valid, user mode
  g0[1] = lds_off;                                     // lds_addr (bytes)
  g0[2] = (unsigned)(ga & 0xFFFFFFFFull);              // global_addr[31:0]
  g0[3] = (unsigned)((ga >> 32) & 0x1FFFFFFull)        // global_addr[56:32]
          | (2u << 30);                                // type = 2 ("image")

  i32x8 g1;
  g1[0] = (int)(2u << 16);        // workgroup_mask=0 | data_size=2 (4 bytes)
  g1[1] = (int)(128u << 16);      // abar_addr=0 | tensor_dim0[15:0]=128
  g1[2] = (int)(nr << 16);        // tensor_dim0[31:16]=0 | tensor_dim1[15:0]=nrows
  g1[3] = (int)(128u << 16);      // tensor_dim1[31:16]=0 | tile_dim0=128
  g1[4] = (int)nr;                // tile_dim1=nrows | tile_dim2=0 (2D)
  g1[5] = 128;                    // tensor_dim0_stride[31:0]=128
  g1[6] = 0;                      // stride0[47:32]=0 | tensor_dim1_stride lo=0
  g1[7] = 0;                      // tensor_dim1_stride hi=0

  i32x4 gz4 = {0, 0, 0, 0};             // groups 2/3 unused (<=2D tensor)
  i32x8 gz8 = {0, 0, 0, 0, 0, 0, 0, 0}; // extra descriptor words (unused)
  __builtin_amdgcn_tensor_load_to_lds(g0, g1, gz4, gz4, gz8, 0);
}
#else
#define HAVE_TDM 0
#endif

// 3x3 local-max test on raw values (monotonic sigmoid => same decision/order)
__device__ __forceinline__ bool survives(const float* tile, int y, int x, int tr, float v) {
  bool keep = true;
#pragma unroll
  for (int dy = -1; dy <= 1; ++dy) {
    int yy = y + dy;
    if ((unsigned)yy >= (unsigned)H_) continue;
#pragma unroll
    for (int dx = -1; dx <= 1; ++dx) {
      if (dy == 0 && dx == 0) continue;
      int xx = x + dx;
      if ((unsigned)xx >= (unsigned)W_) continue;
      keep = keep && (v >= tile[(tr + dy) * W_ + xx]);
    }
  }
  return keep;
}

// ---------- K1: per-(batch,channel) survivor histogram over key top-12 bits ----------
__global__ __launch_bounds__(256) void k_hist(const float* __restrict__ hm,
                                              unsigned* __restrict__ ghist) {
  __shared__ float tile[34 * W_];
  __shared__ unsigned hist[NBINS];
  const int tid = threadIdx.x;
  const int bc = blockIdx.x;
  const int b = bc / C_;
  const float* img = hm + (size_t)bc * HW_;

  for (int i = tid; i < NBINS; i += 256) hist[i] = 0;
  __syncthreads();

  for (int chunk = 0; chunk < 4; ++chunk) {
    int r0 = chunk * 32;
    stage_tile_async(tile, img, r0, tid);
    for (int e = tid; e < 32 * W_; e += 256) {
      int ly = e >> 7, x = e & (W_ - 1);
      int y = r0 + ly, tr = ly + 1;
      float v = tile[tr * W_ + x];
      if (survives(tile, y, x, tr, v))
        atomicAdd(&hist[flip_key(v) >> 20], 1u);
    }
    __syncthreads();
  }

  unsigned* gh = ghist + (size_t)b * NBINS;
  for (int i = tid; i < NBINS; i += 256) {
    unsigned h = hist[i];
    if (h) atomicAdd(&gh[i], h);
  }
}

// ---------- K2: coarse threshold (largest prefix with cum-count >= K) ----------
__global__ void k_thresh(const unsigned* __restrict__ ghist, unsigned* __restrict__ P) {
  int b = threadIdx.x;
  if (b >= B_) return;
  const unsigned* h = ghist + (size_t)b * NBINS;
  unsigned acc = 0;
  int p = 0;
  for (int i = NBINS - 1; i >= 0; --i) {
    acc += h[i];
    if (acc >= (unsigned)K_) { p = i; break; }
  }
  P[b] = (unsigned)p;
}

// ---------- K3: collect candidates with key prefix >= P (TDM-staged tiles) ----------
__global__ __launch_bounds__(256) void k_collect(const float* __restrict__ hm,
                                                 const unsigned* __restrict__ P,
                                                 unsigned* __restrict__ cnt,
                                                 unsigned long long* __restrict__ cand) {
  __shared__ float tile[34 * W_];
  __shared__ unsigned sP;
  const int tid = threadIdx.x;
  const int bc = blockIdx.x;
  const int b = bc / C_;
  const float* img = hm + (size_t)bc * HW_;
  if (tid == 0) sP = P[b];
  __syncthreads();
  const unsigned thr = sP;
  unsigned long long* cb = cand + (size_t)b * CANDMAX;
  unsigned* cc = cnt + b;
  const unsigned baseIdx = (unsigned)((bc % C_) * HW_);

  for (int chunk = 0; chunk < 4; ++chunk) {
    int r0 = chunk * 32;
#if HAVE_TDM
    // Clamp halo to valid rows; tile row t <-> global row (r0-1+t).
    int grow0 = (r0 - 1 < 0) ? 0 : (r0 - 1);
    int grow1 = (r0 + 32 > H_ - 1) ? (H_ - 1) : (r0 + 32);
    int nrows = grow1 - grow0 + 1;
    int ldsrow = grow0 - (r0 - 1);
    if (tid == 0) {
      tdm_load_rows(&tile[ldsrow * W_], img + (size_t)grow0 * W_, nrows);
      __builtin_amdgcn_s_wait_tensorcnt(0);
    }
    __syncthreads();
#else
    stage_tile_async(tile, img, r0, tid);
#endif
    for (int e = tid; e < 32 * W_; e += 256) {
      int ly = e >> 7, x = e & (W_ - 1);
      int y = r0 + ly, tr = ly + 1;
      float v = tile[tr * W_ + x];
      if (survives(tile, y, x, tr, v)) {
        unsigned key = flip_key(v);
        if ((key >> 20) >= thr) {
          unsigned idx = baseIdx + (unsigned)(y * W_ + x);
          unsigned pos = atomicAdd(cc, 1u);
          if (pos < CANDMAX)
            cb[pos] = ((unsigned long long)key << 32) | (unsigned)(~idx);
        }
      }
    }
    __syncthreads();
  }
}

// ---------- K4: bitonic sort candidates, emit top-100 detections ----------
__global__ __launch_bounds__(1024) void k_emit(const unsigned long long* __restrict__ cand,
                                               const unsigned* __restrict__ cnt,
                                               const float* __restrict__ bbox,
                                               const float* __restrict__ off,
                                               const int* __restrict__ image_id,
                                               float* __restrict__ out) {
  __shared__ unsigned long long s[CANDMAX];
  const int b = blockIdx.x, tid = threadIdx.x;
  unsigned n = cnt[b];
  if (n > CANDMAX) n = CANDMAX;
  const unsigned long long* cb = cand + (size_t)b * CANDMAX;
  for (int i = tid; i < CANDMAX; i += 1024)
    s[i] = (i < (int)n) ? cb[i] : 0ull;
  __syncthreads();

  // descending bitonic sort over 64-bit composite (key desc, index asc on ties)
  for (int k = 2; k <= CANDMAX; k <<= 1) {
    for (int j = k >> 1; j > 0; j >>= 1) {
      for (int i = tid; i < CANDMAX; i += 1024) {
        int ixj = i ^ j;
        if (ixj > i) {
          unsigned long long a = s[i], c = s[ixj];
          bool desc = ((i & k) == 0);
          if (desc ? (a < c) : (a > c)) { s[i] = c; s[ixj] = a; }
        }
      }
      __syncthreads();
    }
  }

  if (tid < K_) {
    unsigned long long e = s[tid];
    unsigned key = (unsigned)(e >> 32);
    unsigned idx = ~((unsigned)e);
    float score, cls, xs, ys, ox, oy, ww, hh;
    if (tid < (int)n) {
      float raw = unflip_key(key);
      score = 1.0f / (1.0f + expf(-raw));
      unsigned sp = idx % HW_;
      unsigned cc = idx / HW_;
      cls = (float)cc;
      xs = (float)(sp & (W_ - 1));
      ys = (float)(sp >> 7);
      const float* op = off + (size_t)b * 2 * HW_;
      ox = op[sp];
      oy = op[HW_ + sp];
      const float* wp = bbox + (size_t)b * 2 * HW_;
      ww = wp[sp];
      hh = wp[HW_ + sp];
    } else {
      score = 0.f; cls = 0.f; xs = ys = ox = oy = ww = hh = 0.f;
    }
    float cx = xs + ox, cy = ys + oy;
    float* o = out + ((size_t)b * K_ + tid) * 7;
    o[0] = (float)image_id[b];
    o[1] = (cx - ww * 0.5f) * DOWN_;
    o[2] = (cy - hh * 0.5f) * DOWN_;
    o[3] = (cx + ww * 0.5f) * DOWN_;
    o[4] = (cy + hh * 0.5f) * DOWN_;
    o[5] = score;
    o[6] = cls;
  }
}

extern "C" void kernel_launch(void* const* d_in, const int* in_sizes, int n_in,
                              void* d_out, int out_size, void* d_ws, size_t ws_size,
                              hipStream_t stream) {
  const float* hm   = (const float*)d_in[0];  // (B,C,H,W)
  const float* bbox = (const float*)d_in[1];  // (B,2,H,W)
  const float* off  = (const float*)d_in[2];  // (B,2,H,W)
  const int*   iid  = (const int*)d_in[3];    // (B)
  float* out = (float*)d_out;                 // (B,K,7)

  unsigned* ws = (unsigned*)d_ws;
  unsigned* ghist = ws;                                   // B*4096 u32
  unsigned* P     = ws + B_ * NBINS;                      // 16 u32
  unsigned* cnt   = P + 16;                               // 16 u32 (padded to 64)
  unsigned long long* cand =
      (unsigned long long*)(ws + B_ * NBINS + 64);        // B*2048 u64

  size_t zero_bytes = (size_t)(B_ * NBINS + 64) * sizeof(unsigned);
  (void)hipMemsetAsync(d_ws, 0, zero_bytes, stream);

  k_hist   <<<B_ * C_, 256,  0, stream>>>(hm, ghist);
  k_thresh <<<1,       B_,   0, stream>>>(ghist, P);
  k_collect<<<B_ * C_, 256,  0, stream>>>(hm, P, cnt, cand);
  k_emit   <<<B_,      1024, 0, stream>>>(cand, cnt, bbox, off, iid, out);
}